// EncoderSpin_9766755631279
// MI455X (gfx1250) — compile-verified
//
#include <hip/hip_runtime.h>

typedef __attribute__((ext_vector_type(2))) float v2f;
typedef __attribute__((ext_vector_type(8))) float v8f;

#define TPB 256

// ---------------------------------------------------------------------------
// ws layout (floats): dinv[N] (deg during accumulation), s[N], h[N*32], g[N*32]
// ---------------------------------------------------------------------------

// deg[i] = 1.0 (self loop contribution)
__global__ void gcnv_init_deg(float* __restrict__ deg, int n) {
    int i = blockIdx.x * blockDim.x + threadIdx.x;
    if (i < n) deg[i] = 1.0f;
}

// deg[dst[e]] += 1
__global__ void gcnv_degree(const long long* __restrict__ dst,
                            float* __restrict__ deg, int ne) {
    int e = blockIdx.x * blockDim.x + threadIdx.x;
    if (e < ne) atomicAdd(&deg[(int)dst[e]], 1.0f);
}

// dinv = rsqrt(deg) (in place); s[i] = x[i]*dinv^2 (self loop term of layer 1)
__global__ void gcnv_dinv_sinit(float* __restrict__ dinv,
                                const float* __restrict__ x,
                                float* __restrict__ s, int n) {
    int i = blockIdx.x * blockDim.x + threadIdx.x;
    if (i < n) {
        float di = rsqrtf(dinv[i]);
        dinv[i] = di;
        s[i] = x[i] * di * di;
    }
}

// s[dst] += x[src] * dinv[src] * dinv[dst]   (layer-1 scalar aggregation)
__global__ void gcnv_agg1(const long long* __restrict__ src,
                          const long long* __restrict__ dst,
                          const float* __restrict__ x,
                          const float* __restrict__ dinv,
                          float* __restrict__ s, int ne) {
    int e = blockIdx.x * blockDim.x + threadIdx.x;
    if (e < ne) {
        int si = (int)src[e], di = (int)dst[e];
        atomicAdd(&s[di], x[si] * dinv[si] * dinv[di]);
    }
}

// h[i][f] = relu(s[i]*W1[f] + b1[f]);  g[i][f] = h[i][f]*dinv[i]^2 (self loop)
__global__ void gcnv_h_ginit(const float* __restrict__ s,
                             const float* __restrict__ W1,
                             const float* __restrict__ b1,
                             const float* __restrict__ dinv,
                             float* __restrict__ h,
                             float* __restrict__ g, int n) {
    int t = blockIdx.x * blockDim.x + threadIdx.x;
    if (t < n * 32) {
        int i = t >> 5, f = t & 31;
        float hv = fmaxf(s[i] * W1[f] + b1[f], 0.0f);
        h[t] = hv;
        float di = dinv[i];
        g[t] = hv * di * di;
    }
}

// g[dst][:] += h[src][:] * norm   — 8 threads per edge, float4 each
__global__ void gcnv_agg2(const long long* __restrict__ src,
                          const long long* __restrict__ dst,
                          const float* __restrict__ dinv,
                          const float* __restrict__ h,
                          float* __restrict__ g, int ne) {
    int t = blockIdx.x * blockDim.x + threadIdx.x;
    if (t < ne * 8) {
        int e = t >> 3, q = t & 7;
        int si = (int)src[e], di = (int)dst[e];
        float norm = dinv[si] * dinv[di];
        float4 hv = ((const float4*)(h + (size_t)si * 32))[q];
        float* gb = g + (size_t)di * 32 + q * 4;
        atomicAdd(gb + 0, hv.x * norm);
        atomicAdd(gb + 1, hv.y * norm);
        atomicAdd(gb + 2, hv.z * norm);
        atomicAdd(gb + 3, hv.w * norm);
    }
}

// One wave per 16-node tile: [16x32] @ [32x16] (+bias), twice (mu, logvar),
// via 8+8 chained V_WMMA_F32_16X16X4_F32.
// Fragment layouts (wave32):
//   A 16x4 f32:  lane l -> row m=l&15, k0=(l>>4)*2; a[j] = A[m][k0+j]
//   B 4x16 f32:  lane l -> col n=l&15, k0=(l>>4)*2; b[j] = B[k0+j][n]
//   C/D 16x16:   vgpr r, lane l -> m = r + (l>>4)*8, n = l&15
__global__ __launch_bounds__(TPB) void gcnv_out_wmma(
    const float* __restrict__ g,
    const float* __restrict__ Wmu, const float* __restrict__ bmu,
    const float* __restrict__ Wlv, const float* __restrict__ blv,
    float* __restrict__ out_mu, float* __restrict__ out_lv, int ntiles) {
    int wave = (int)((blockIdx.x * blockDim.x + threadIdx.x) >> 5);
    int lane = threadIdx.x & 31;
    if (wave >= ntiles) return;  // uniform per wave: EXEC all-1s inside

    int m = lane & 15;       // row within tile (A), col (B / C / D)
    int half = lane >> 4;

    // A fragments: rows of g for this tile
    v2f a[8];
    const float* grow = g + ((size_t)wave * 16 + m) * 32;
#pragma unroll
    for (int kb = 0; kb < 8; ++kb) {
        int k0 = kb * 4 + half * 2;
        a[kb].x = grow[k0];
        a[kb].y = grow[k0 + 1];
    }
    // B fragments: the two weight matrices ([32,16] row-major)
    v2f wm[8], wl[8];
#pragma unroll
    for (int kb = 0; kb < 8; ++kb) {
        int k0 = kb * 4 + half * 2;
        wm[kb].x = Wmu[(k0 + 0) * 16 + m];
        wm[kb].y = Wmu[(k0 + 1) * 16 + m];
        wl[kb].x = Wlv[(k0 + 0) * 16 + m];
        wl[kb].y = Wlv[(k0 + 1) * 16 + m];
    }
    // C init = bias (depends only on column n == m here)
    v8f cmu, clv;
    float bm = bmu[m], bl = blv[m];
#pragma unroll
    for (int r = 0; r < 8; ++r) { cmu[r] = bm; clv[r] = bl; }

#pragma unroll
    for (int kb = 0; kb < 8; ++kb) {
        cmu = __builtin_amdgcn_wmma_f32_16x16x4_f32(false, a[kb], false, wm[kb],
                                                    (short)0, cmu, false, false);
        clv = __builtin_amdgcn_wmma_f32_16x16x4_f32(false, a[kb], false, wl[kb],
                                                    (short)0, clv, false, false);
    }
#pragma unroll
    for (int r = 0; r < 8; ++r) {
        size_t row = (size_t)wave * 16 + r + half * 8;
        out_mu[row * 16 + m] = cmu[r];
        out_lv[row * 16 + m] = clv[r];
    }
}

// Scalar fallback for a possible non-multiple-of-16 tail (N=100000 -> unused)
__global__ void gcnv_out_tail(const float* __restrict__ g,
                              const float* __restrict__ Wmu, const float* __restrict__ bmu,
                              const float* __restrict__ Wlv, const float* __restrict__ blv,
                              float* __restrict__ out_mu, float* __restrict__ out_lv,
                              int start, int n) {
    int t = blockIdx.x * blockDim.x + threadIdx.x;
    int i = start + (t >> 4), f = t & 15;
    if (i < n) {
        const float* gi = g + (size_t)i * 32;
        float accm = bmu[f], accl = blv[f];
        for (int k = 0; k < 32; ++k) {
            float gv = gi[k];
            accm += gv * Wmu[k * 16 + f];
            accl += gv * Wlv[k * 16 + f];
        }
        out_mu[(size_t)i * 16 + f] = accm;
        out_lv[(size_t)i * 16 + f] = accl;
    }
}

extern "C" void kernel_launch(void* const* d_in, const int* in_sizes, int n_in,
                              void* d_out, int out_size, void* d_ws, size_t ws_size,
                              hipStream_t stream) {
    const float* x = (const float*)d_in[0];
    const long long* ei = (const long long*)d_in[1];
    const float* W1  = (const float*)d_in[2];
    const float* b1  = (const float*)d_in[3];
    const float* Wmu = (const float*)d_in[4];
    const float* bmu = (const float*)d_in[5];
    const float* Wlv = (const float*)d_in[6];
    const float* blv = (const float*)d_in[7];

    const int n  = in_sizes[0];       // nodes (x is [N,1])
    const int ne = in_sizes[1] / 2;   // edges (edge_index is [2,E])
    const long long* src = ei;
    const long long* dst = ei + ne;

    float* ws   = (float*)d_ws;
    float* dinv = ws;                       // [N] (deg during accumulation)
    float* s    = ws + n;                   // [N]
    float* h    = ws + 2 * (size_t)n;       // [N*32]
    float* g    = h + (size_t)n * 32;       // [N*32]

    float* out_mu = (float*)d_out;
    float* out_lv = (float*)d_out + (size_t)n * 16;

    int gb_n   = (n + TPB - 1) / TPB;
    int gb_e   = (ne + TPB - 1) / TPB;
    int gb_nf  = (n * 32 + TPB - 1) / TPB;
    int gb_e8  = (int)(((long long)ne * 8 + TPB - 1) / TPB);

    gcnv_init_deg<<<gb_n, TPB, 0, stream>>>(dinv, n);
    gcnv_degree<<<gb_e, TPB, 0, stream>>>(dst, dinv, ne);
    gcnv_dinv_sinit<<<gb_n, TPB, 0, stream>>>(dinv, x, s, n);
    gcnv_agg1<<<gb_e, TPB, 0, stream>>>(src, dst, x, dinv, s, ne);
    gcnv_h_ginit<<<gb_nf, TPB, 0, stream>>>(s, W1, b1, dinv, h, g, n);
    gcnv_agg2<<<gb_e8, TPB, 0, stream>>>(src, dst, dinv, h, g, ne);

    int ntiles = n / 16;
    if (ntiles > 0) {
        int gb_w = (ntiles * 32 + TPB - 1) / TPB;  // 8 waves per block
        gcnv_out_wmma<<<gb_w, TPB, 0, stream>>>(g, Wmu, bmu, Wlv, blv,
                                                out_mu, out_lv, ntiles);
    }
    int rem = n - ntiles * 16;
    if (rem > 0) {
        int gb_t = (rem * 16 + TPB - 1) / TPB;
        gcnv_out_tail<<<gb_t, TPB, 0, stream>>>(g, Wmu, bmu, Wlv, blv,
                                                out_mu, out_lv, ntiles * 16, n);
    }
}